// EnokeeEncoder_65953517797480
// MI455X (gfx1250) — compile-verified
//
#include <hip/hip_runtime.h>
#include <hip/hip_bf16.h>

// ---------------------------------------------------------------------------
// Shapes (compile-time constants from the reference)
// ---------------------------------------------------------------------------
#define Bq   16
#define Sq   512
#define Mq   64
#define Lq   32
#define Dq   1024
#define Rq   128
#define Eq   100000
#define LN_EPS 1e-5f

typedef __attribute__((ext_vector_type(2))) float v2f;
typedef __attribute__((ext_vector_type(8))) float v8f;

// ---------------------------------------------------------------------------
// Kernel 1: masked mention pooling + LayerNorm + x@w1  ->  h [B*M, R] f32
// One block per (b, m); 128 threads. Negligible vs the final GEMM.
// ---------------------------------------------------------------------------
__global__ void pool_ln_w1_kernel(const int* __restrict__ input_ids,
                                  const int* __restrict__ epid,
                                  const int* __restrict__ eam,
                                  const float* __restrict__ embed,
                                  const float* __restrict__ gamma,
                                  const float* __restrict__ beta,
                                  const float* __restrict__ w1,
                                  float* __restrict__ h) {
    const int bm  = blockIdx.x;          // 0 .. B*M-1
    const int b   = bm / Mq;
    const int tid = threadIdx.x;         // 0 .. 127

    __shared__ float xs[Dq];             // normalized pooled vector
    __shared__ float rs[128], rss[128];  // reduction scratch

    const int am = eam[bm];

    float acc[8];
#pragma unroll
    for (int j = 0; j < 8; ++j) acc[j] = 0.0f;

    for (int l = 0; l < Lq; ++l) {
        const bool valid = (epid[bm * Lq + l] != -1) && (am != 0);
        if (valid) {
            const int tok = input_ids[b * Sq + l];
            const float* e = embed + (size_t)tok * Dq;
#pragma unroll
            for (int j = 0; j < 8; ++j) acc[j] += e[tid + j * 128];
        }
    }
    const float invL = 1.0f / (float)Lq;
#pragma unroll
    for (int j = 0; j < 8; ++j) acc[j] *= invL;

    float s = 0.0f, ss = 0.0f;
#pragma unroll
    for (int j = 0; j < 8; ++j) { s += acc[j]; ss += acc[j] * acc[j]; }
    rs[tid] = s; rss[tid] = ss;
    __syncthreads();
    for (int off = 64; off > 0; off >>= 1) {
        if (tid < off) { rs[tid] += rs[tid + off]; rss[tid] += rss[tid + off]; }
        __syncthreads();
    }
    const float mu   = rs[0]  * (1.0f / (float)Dq);
    const float var  = rss[0] * (1.0f / (float)Dq) - mu * mu;
    const float rstd = rsqrtf(var + LN_EPS);
    __syncthreads();

#pragma unroll
    for (int j = 0; j < 8; ++j) {
        const int d = tid + j * 128;
        xs[d] = (acc[j] - mu) * rstd * gamma[d] + beta[d];
    }
    __syncthreads();

    float hr = 0.0f;
    for (int d = 0; d < Dq; ++d) hr = fmaf(xs[d], w1[d * Rq + tid], hr);
    h[(size_t)bm * Rq + tid] = hr;
}

// ---------------------------------------------------------------------------
// Kernel 2: out[1024, 100000] = h[1024,128] @ w2[128,100000] + b2
//
// WMMA f32 16x16x4. Block = 128 threads = 4 waves, computing a 64M x 64N
// output tile. The block's 64x128 A panel (32 KB) is staged in LDS once
// (b128 loads/stores). Each wave owns one 16-wide N tile and FOUR M tiles:
// per K step it loads one B fragment (2 x b32 from L2-resident w2) and
// feeds 4 WMMAs, quartering L2 B-traffic (3.2 GB -> 0.8 GB) so it balances
// the 410 MB HBM output stream. 128 v_wmma per wave, fully unrolled.
// ---------------------------------------------------------------------------
__global__ void gemm_wmma_kernel(const float* __restrict__ h,
                                 const float* __restrict__ w2,
                                 const float* __restrict__ b2,
                                 float* __restrict__ out) {
    __shared__ float As[64 * Rq];        // 32 KB: 64 rows x 128 K

    const int mblk = blockIdx.y;         // 0 .. 15 (64-row M blocks)
    const int tid  = threadIdx.x;

    // Stage 64x128 A panel: 8192 floats = 2048 float4, 16 per thread.
    {
        const float4* hA4 = (const float4*)(h + (size_t)mblk * 64 * Rq);
        float4*       As4 = (float4*)As;
#pragma unroll
        for (int i = tid; i < 2048; i += 128) As4[i] = hA4[i];
    }
    __syncthreads();

    const int wave  = tid >> 5;
    const int lane  = tid & 31;
    const int ntile = blockIdx.x * 4 + wave;
    if (ntile >= Eq / 16) return;        // 6250 exact N tiles; tail guard

    const int n0    = ntile * 16;
    const int nIdx  = lane & 15;         // column within 16-wide N tile
    const int row   = lane & 15;         // A row within a 16-row M tile
    const int khalf = (lane >> 4) << 1;  // 0 for lanes 0-15, 2 for lanes 16-31

    v8f acc0 = {}, acc1 = {}, acc2 = {}, acc3 = {};

    const float* Bp = w2 + (size_t)khalf * Eq + (n0 + nIdx);
    // Per-M-tile A base pointers into LDS (16 rows apart).
    const float* Ap0 = As + (row +  0) * Rq + khalf;
    const float* Ap1 = As + (row + 16) * Rq + khalf;
    const float* Ap2 = As + (row + 32) * Rq + khalf;
    const float* Ap3 = As + (row + 48) * Rq + khalf;

#pragma unroll
    for (int k0 = 0; k0 < Rq; k0 += 4) {
        // One B fragment per K step, reused by 4 WMMAs.
        v2f bb;
        bb.x = Bp[(size_t)k0 * Eq];
        bb.y = Bp[(size_t)(k0 + 1) * Eq];

        v2f a0, a1, a2, a3;
        a0.x = Ap0[k0]; a0.y = Ap0[k0 + 1];
        a1.x = Ap1[k0]; a1.y = Ap1[k0 + 1];
        a2.x = Ap2[k0]; a2.y = Ap2[k0 + 1];
        a3.x = Ap3[k0]; a3.y = Ap3[k0 + 1];

        acc0 = __builtin_amdgcn_wmma_f32_16x16x4_f32(false, a0, false, bb,
                                                     (short)0, acc0, false, false);
        acc1 = __builtin_amdgcn_wmma_f32_16x16x4_f32(false, a1, false, bb,
                                                     (short)0, acc1, false, false);
        acc2 = __builtin_amdgcn_wmma_f32_16x16x4_f32(false, a2, false, bb,
                                                     (short)0, acc2, false, false);
        acc3 = __builtin_amdgcn_wmma_f32_16x16x4_f32(false, a3, false, bb,
                                                     (short)0, acc3, false, false);
    }

    // C/D layout: VGPR j -> M = j (lanes 0-15) or j+8 (lanes 16-31).
    const float bias = b2[n0 + nIdx];
    const int   rsub = (lane >> 4) << 3;                // 0 or 8
    const size_t col = (size_t)(n0 + nIdx);
    float* o0 = out + (size_t)(mblk * 64 +  0 + rsub) * Eq + col;
    float* o1 = out + (size_t)(mblk * 64 + 16 + rsub) * Eq + col;
    float* o2 = out + (size_t)(mblk * 64 + 32 + rsub) * Eq + col;
    float* o3 = out + (size_t)(mblk * 64 + 48 + rsub) * Eq + col;
#pragma unroll
    for (int j = 0; j < 8; ++j) {
        __builtin_nontemporal_store(acc0[j] + bias, o0 + (size_t)j * Eq);
        __builtin_nontemporal_store(acc1[j] + bias, o1 + (size_t)j * Eq);
        __builtin_nontemporal_store(acc2[j] + bias, o2 + (size_t)j * Eq);
        __builtin_nontemporal_store(acc3[j] + bias, o3 + (size_t)j * Eq);
    }
}

// ---------------------------------------------------------------------------
// Launch
// ---------------------------------------------------------------------------
extern "C" void kernel_launch(void* const* d_in, const int* in_sizes, int n_in,
                              void* d_out, int out_size, void* d_ws, size_t ws_size,
                              hipStream_t stream) {
    const int*   input_ids = (const int*)  d_in[0];   // [B,S]
    const int*   epid      = (const int*)  d_in[1];   // [B,M,L]
    const int*   eam       = (const int*)  d_in[2];   // [B,M]
    const float* embed     = (const float*)d_in[3];   // [V,D]
    const float* ln_gamma  = (const float*)d_in[4];   // [D]
    const float* ln_beta   = (const float*)d_in[5];   // [D]
    const float* w1        = (const float*)d_in[6];   // [D,R]
    const float* w2        = (const float*)d_in[7];   // [R,E]
    const float* b2        = (const float*)d_in[8];   // [E]
    float*       out       = (float*)d_out;           // [B,M,E]

    float* h = (float*)d_ws;                          // [B*M, R] = 512 KB scratch

    // Stage 1: pooling + LN + w1 GEMV
    pool_ln_w1_kernel<<<Bq * Mq, 128, 0, stream>>>(
        input_ids, epid, eam, embed, ln_gamma, ln_beta, w1, h);

    // Stage 2: WMMA GEMM vs w2 (dominant work)
    dim3 grid((Eq / 16 + 3) / 4, 16);                 // (1563, 16), 64Mx64N / block
    gemm_wmma_kernel<<<grid, 128, 0, stream>>>(h, w2, b2, out);
}